// Attender_12927851561399
// MI455X (gfx1250) — compile-verified
//
#include <hip/hip_runtime.h>
#include <stdint.h>

// MI455X / gfx1250, wave32. fp32 WMMA 16x16x4 for both GEMM stages.
// Roofline: 272 MB HBM traffic @ 23.3 TB/s ~ 12us; compute negligible.

typedef float v2f __attribute__((ext_vector_type(2)));
typedef float v8f __attribute__((ext_vector_type(8)));

#define S_DIM 2048
#define B_DIM 16
#define D_DIM 2048

// ---------------------------------------------------------------------------
// Stage 1: altered[16, D] = state[16, D] @ W^T + bias
// One wave computes one 16x16 output tile, looping K over D in chunks of 4.
// f32 WMMA operand layout (ISA 7.12.2): lanes 0-15 hold K={0,1}, lanes 16-31
// hold K={2,3}; both A and B are one float2 load per lane.
// ---------------------------------------------------------------------------
__global__ __launch_bounds__(32) void stage1_linear(
    const float* __restrict__ state,   // [16, D]
    const float* __restrict__ W,       // [D, D] row-major (out, in)
    const float* __restrict__ bias,    // [D]
    float* __restrict__ altered)       // [16, D]
{
  const int tile = blockIdx.x;            // 0..127 column tiles
  const int lane = threadIdx.x & 31;
  const int n    = lane & 15;             // column-in-tile; also A row m
  const int koff = (lane >> 4) * 2;       // hi half-wave carries K+2,K+3
  const int col  = tile * 16 + n;

  const float* aPtr = state + (size_t)n   * D_DIM + koff; // A[m, k0+koff..+1]
  const float* bPtr = W     + (size_t)col * D_DIM + koff; // B[k,n] = W[col,k]

  v8f acc0 = {}, acc1 = {};               // split accumulators: 2 WMMA chains
  #pragma unroll 4
  for (int k0 = 0; k0 < D_DIM; k0 += 8) {
    v2f a0 = *(const v2f*)(aPtr + k0);
    v2f b0 = *(const v2f*)(bPtr + k0);
    acc0 = __builtin_amdgcn_wmma_f32_16x16x4_f32(false, a0, false, b0,
                                                 (short)0, acc0, false, false);
    v2f a1 = *(const v2f*)(aPtr + k0 + 4);
    v2f b1 = *(const v2f*)(bPtr + k0 + 4);
    acc1 = __builtin_amdgcn_wmma_f32_16x16x4_f32(false, a1, false, b1,
                                                 (short)0, acc1, false, false);
  }
  v8f acc = acc0 + acc1;

  // C/D layout: VGPR r -> row m = r + 8*(lane>=16), column n = lane&15.
  const float bv    = bias[col];          // bias depends only on column
  const int   mbase = (lane >> 4) * 8;
  #pragma unroll
  for (int r = 0; r < 8; ++r)
    altered[(size_t)(mbase + r) * D_DIM + col] = acc[r] + bv;
}

// ---------------------------------------------------------------------------
// Stage 2: out[b, s] = sum_d altered[b, d] * enc[s, b, d]
// Per s:  WMMA(altered[16,4], enc_chunk^T[4,16]) -> 16x16; diagonal is the
// batched dot for all 16 b at once. enc streamed HBM->LDS with async b128
// copies (coalesced), double-buffered; WMMA operands read from LDS.
// ---------------------------------------------------------------------------
#define KT         32                    // K-floats staged per stage
#define ROW_STRIDE 36                    // KT + 4 pad: 16B-aligned rows, 0 bank conflicts
#define NSTAGE     (D_DIM / KT)          // 64 stages

__device__ __forceinline__ void async_b128_to_lds(uint32_t lds_off,
                                                  const float* gptr) {
  asm volatile("global_load_async_to_lds_b128 %0, %1, off"
               :: "v"(lds_off), "v"(gptr) : "memory");
}

__global__ __launch_bounds__(256) void stage2_attend(
    const float* __restrict__ enc,      // [S, 16, D]
    const float* __restrict__ altered,  // [16, D]
    float* __restrict__ out)            // [16, S]
{
  __shared__ float s_enc[2][128 * ROW_STRIDE];  // 8 s * 16 b rows of KT floats
  __shared__ float s_alt[2][16 * ROW_STRIDE];   // 16 b rows of KT floats

  const int t    = threadIdx.x;
  const int w    = t >> 5;               // wave id -> local s index
  const int lane = t & 31;
  const int n    = lane & 15;            // column n == batch b; also A row m
  const int hi   = lane >> 4;
  const int s0   = blockIdx.x * 8;

  // --- async prefetch of one KT-wide stage into LDS buffer `buf` ---
  auto prefetch = [&](int js, int buf) {
    const int d0 = js * KT;
    // enc slab: 128 rows (8 s x 16 b) x KT floats = 1024 b128 chunks
    #pragma unroll
    for (int j = 0; j < 4; ++j) {
      const int c   = t + j * 256;
      const int row = c >> 3;            // 8 chunks per row
      const int pos = (c & 7) * 4;
      const int si  = row >> 4, bi = row & 15;
      const float* g = enc +
          ((size_t)((s0 + si) * B_DIM + bi)) * D_DIM + d0 + pos;
      async_b128_to_lds(
          (uint32_t)(uintptr_t)&s_enc[buf][row * ROW_STRIDE + pos], g);
    }
    // altered slab: 16 rows x KT floats = 128 b128 chunks (waves 0-3 only;
    // wave-uniform predicate, async instr issues with partial EXEC legally)
    if (w < 4) {
      const int row = t >> 3;
      const int pos = (t & 7) * 4;
      async_b128_to_lds(
          (uint32_t)(uintptr_t)&s_alt[buf][row * ROW_STRIDE + pos],
          altered + (size_t)row * D_DIM + d0 + pos);
    }
  };

  const int rowA  = n * ROW_STRIDE;               // altered row m = n
  const int rowB  = (w * 16 + n) * ROW_STRIDE;    // enc row for (s_local, b=n)
  const int kbase = 2 * hi;                       // hi half-wave: K+2,K+3

  v8f acc0 = {}, acc1 = {};
  prefetch(0, 0);

  for (int js = 0; js < NSTAGE; ++js) {
    // my stage-js async loads landed in LDS:
    asm volatile("s_wait_asynccnt 0x0" ::: "memory");
    // everyone's stage-js landed AND everyone finished reading buf ~js:
    __syncthreads();
    if (js + 1 < NSTAGE) prefetch(js + 1, (js + 1) & 1);   // overlaps compute

    const float* eb = &s_enc[js & 1][0];
    const float* ab = &s_alt[js & 1][0];
    #pragma unroll
    for (int kc = 0; kc < KT / 4; kc += 2) {
      const int kk0 = kc * 4 + kbase;
      v2f a0 = *(const v2f*)(ab + rowA + kk0);    // ds_load_b64
      v2f b0 = *(const v2f*)(eb + rowB + kk0);
      acc0 = __builtin_amdgcn_wmma_f32_16x16x4_f32(false, a0, false, b0,
                                                   (short)0, acc0, false, false);
      const int kk1 = kk0 + 4;
      v2f a1 = *(const v2f*)(ab + rowA + kk1);
      v2f b1 = *(const v2f*)(eb + rowB + kk1);
      acc1 = __builtin_amdgcn_wmma_f32_16x16x4_f32(false, a1, false, b1,
                                                   (short)0, acc1, false, false);
    }
  }

  // Diagonal extraction: VGPR r / lane -> (m = r + 8*hi, n); keep m == n.
  v8f acc = acc0 + acc1;
  const int s = s0 + w;
  #pragma unroll
  for (int r = 0; r < 8; ++r) {
    const int m = r + hi * 8;
    if (m == n) out[(size_t)m * S_DIM + s] = acc[r];
  }
}

// ---------------------------------------------------------------------------
extern "C" void kernel_launch(void* const* d_in, const int* in_sizes, int n_in,
                              void* d_out, int out_size, void* d_ws, size_t ws_size,
                              hipStream_t stream) {
  const float* enc   = (const float*)d_in[0];  // [S, B, D]
  const float* state = (const float*)d_in[1];  // [B, D]
  const float* W     = (const float*)d_in[2];  // [D, D]
  const float* bias  = (const float*)d_in[3];  // [D]
  float*       out   = (float*)d_out;          // [B, S]
  float*       alt   = (float*)d_ws;           // scratch: 16*2048 floats

  stage1_linear<<<D_DIM / 16, 32, 0, stream>>>(state, W, bias, alt);
  stage2_attend<<<S_DIM / 8, 256, 0, stream>>>(enc, alt, out);
}